// BoucWenNet_86870008529009
// MI455X (gfx1250) — compile-verified
//
#include <hip/hip_runtime.h>

// Problem constants (B,T) = (4096, 8192)
#define BB      4096
#define TT      8192
#define THREADS 256
#define TILE    2048
#define NTILE   (TT / TILE)      // 4
#define SEG     (TILE / THREADS) // 8
#define KITERS  (TILE / THREADS) // 8 coalesced copy iterations per array
#define LDSN    (TILE + TILE / 64) // padded LDS array length (2080)

// ---- CDNA5 async global->LDS path (guarded so it compiles on any toolchain) ----
#if defined(__AMDGCN__) && __has_builtin(__builtin_amdgcn_global_load_async_to_lds_b32)
#define HAS_ASYNC 1
#else
#define HAS_ASYNC 0
#endif

#if HAS_ASYNC
typedef __attribute__((address_space(1))) int gint_t;
typedef __attribute__((address_space(3))) int lint_t;
__device__ __forceinline__ void async_copy_b32(const float* gsrc, float* ldst) {
  __builtin_amdgcn_global_load_async_to_lds_b32((gint_t*)gsrc, (lint_t*)ldst, 0, 0);
}
#if __has_builtin(__builtin_amdgcn_s_wait_asynccnt)
#define WAIT_ASYNC(n) __builtin_amdgcn_s_wait_asynccnt(n)
#else
#define WAIT_ASYNC(n) asm volatile("s_wait_asynccnt %0" ::"i"(n) : "memory")
#endif
#else
#define WAIT_ASYNC(n)
#endif

__device__ __forceinline__ int pad(int i) { return i + (i >> 6); }

// Stage one tile (x and x_pre) into padded LDS buffers, coalesced.
__device__ __forceinline__ void issue_tile(const float* __restrict__ xg,
                                           const float* __restrict__ pg,
                                           float* sx, float* sp, int tid) {
#pragma unroll
  for (int k = 0; k < KITERS; ++k) {
    const int i  = k * THREADS + tid;
    const int pi = pad(i);
#if HAS_ASYNC
    async_copy_b32(xg + i, sx + pi);
    async_copy_b32(pg + i, sp + pi);
#else
    sx[pi] = xg[i];
    sp[pi] = pg[i];
#endif
  }
}

__global__ __launch_bounds__(THREADS) void boucwen_scan_kernel(
    const float* __restrict__ x, const float* __restrict__ x_pre,
    const float* __restrict__ h_init, const float* __restrict__ w,
    float* __restrict__ out, float* __restrict__ w_out, float* __restrict__ h_out) {
  __shared__ float sX[2][LDSN];
  __shared__ float sP[2][LDSN];
  __shared__ float sA[THREADS];
  __shared__ float sC[THREADS];

  const int tid  = threadIdx.x;
  const int row  = blockIdx.x;
  const long rb  = (long)row * TT;

  const float alpha1 = w[0], alpha2 = w[1], Aw = w[2];
  const float vw = w[3], delta = w[4], bw = w[5];

  float h = h_init[row];  // cross-tile carry (redundant per thread, all equal)

  // Prime the pipeline: tile 0 -> buffer 0
  issue_tile(x + rb, x_pre + rb, sX[0], sP[0], tid);

  for (int t = 0; t < NTILE; ++t) {
    const int cur = t & 1;
    if (t + 1 < NTILE) {
      // Issue next tile into the other buffer, then wait for current tile only.
      issue_tile(x + rb + (long)(t + 1) * TILE, x_pre + rb + (long)(t + 1) * TILE,
                 sX[cur ^ 1], sP[cur ^ 1], tid);
      WAIT_ASYNC(2 * KITERS);   // 16 outstanding = only next tile in flight
    } else {
      WAIT_ASYNC(0);
    }
    __syncthreads();

    // ---- Pass A: per-thread segment -> affine aggregate, all locals in VGPRs
    float av[SEG], cv[SEG], bs[SEG];
    float Ag = 1.0f, Cg = 0.0f;
    const int ib = tid * SEG;
#pragma unroll
    for (int j = 0; j < SEG; ++j) {
      const int pi = pad(ib + j);
      const float xv = sX[cur][pi];
      const float pv = sP[cur][pi];
      const float dx = xv - pv;
      const float e  = __expf(2.0f * dx);
      const float ax = __fdividef(alpha1 * e + alpha2, e + 1.0f);
      const float ad = fabsf(dx);
      const float a  = __fdividef(1.0f, 1.0f + vw * ad);
      const float c  = (Aw * dx + delta * ad) * a;
      av[j] = a; cv[j] = c; bs[j] = ax * xv + bw;
      Ag = a * Ag;
      Cg = a * Cg + c;
    }
    sA[tid] = Ag; sC[tid] = Cg;
    __syncthreads();

    // ---- Hillis-Steele inclusive scan of affine pairs over 256 threads
#pragma unroll
    for (int d = 1; d < THREADS; d <<= 1) {
      float aL = 1.0f, cL = 0.0f;
      const bool has = (tid >= d);
      if (has) { aL = sA[tid - d]; cL = sC[tid - d]; }
      __syncthreads();
      if (has) {
        const float aR = sA[tid], cR = sC[tid];
        sA[tid] = aR * aL;
        sC[tid] = aR * cL + cR;
      }
      __syncthreads();
    }

    // Exclusive prefix for this thread; tile total for carry update.
    float Ae = 1.0f, Ce = 0.0f;
    if (tid > 0) { Ae = sA[tid - 1]; Ce = sC[tid - 1]; }
    const float At = sA[THREADS - 1], Ct = sC[THREADS - 1];

    // ---- Pass B: replay segment from registers, stage out into sX[cur]
    float hs = Ae * h + Ce;
#pragma unroll
    for (int j = 0; j < SEG; ++j) {
      hs = av[j] * hs + cv[j];
      sX[cur][pad(ib + j)] = bs[j] + hs;
    }
    h = At * h + Ct;  // advance carry (same value in every thread)
    __syncthreads();

    // ---- Coalesced store of staged outputs
#pragma unroll
    for (int k = 0; k < KITERS; ++k) {
      const int i = k * THREADS + tid;
      out[rb + (long)t * TILE + i] = sX[cur][pad(i)];
    }
    __syncthreads();  // protect sX[cur] before it is reused as async dest (t+2)
  }

  if (tid == 0) h_out[row] = h;
  if (row == 0 && tid < 6) w_out[tid] = w[tid];
}

extern "C" void kernel_launch(void* const* d_in, const int* in_sizes, int n_in,
                              void* d_out, int out_size, void* d_ws, size_t ws_size,
                              hipStream_t stream) {
  const float* x      = (const float*)d_in[0];
  const float* x_pre  = (const float*)d_in[1];
  const float* h_init = (const float*)d_in[2];
  const float* w      = (const float*)d_in[3];

  float* out   = (float*)d_out;                       // B*T elements
  float* w_out = (float*)d_out + (long)BB * TT;       // 6 elements
  float* h_out = (float*)d_out + (long)BB * TT + 6;   // B elements

  boucwen_scan_kernel<<<BB, THREADS, 0, stream>>>(x, x_pre, h_init, w,
                                                  out, w_out, h_out);
}